// OptimizedKANLayer_55181739819483
// MI455X (gfx1250) — compile-verified
//
#include <hip/hip_runtime.h>
#include <hip/hip_bf16.h>

#define BT     2048   // B*T
#define INDIM  64
#define OUTDIM 64
#define NS     8      // NUM_SPLINES
#define NK     12     // NUM_SPLINES + DEGREE + 1
#define KDIM   256    // INDIM*4 activations
#define EPSV   1e-8f

typedef float v2f __attribute__((ext_vector_type(2)));
typedef float v8f __attribute__((ext_vector_type(8)));

// ---------------------------------------------------------------------------
// Kernel 1: per-(o,i) sort of 12 knots + precompute 1/(k[j+d]-k[j]+eps).
// Fully unrolled bubble sort => static indices => stays in VGPRs.
// ---------------------------------------------------------------------------
__global__ void kan_sort_knots(const float* __restrict__ raw,
                               float* __restrict__ knots,
                               float* __restrict__ invd) {
  int p = blockIdx.x * blockDim.x + threadIdx.x;   // 0..4095 == o*64+i
  if (p >= OUTDIM * INDIM) return;
  float k[NK];
#pragma unroll
  for (int j = 0; j < NK; ++j) k[j] = raw[p * NK + j];
#pragma unroll
  for (int a = 0; a < NK - 1; ++a) {
#pragma unroll
    for (int j = 0; j < NK - 1 - a; ++j) {
      float lo = fminf(k[j], k[j + 1]);
      float hi = fmaxf(k[j], k[j + 1]);
      k[j] = lo; k[j + 1] = hi;
    }
  }
#pragma unroll
  for (int j = 0; j < NK; ++j) knots[p * NK + j] = k[j];
  float* iv = invd + p * 30;
  int idx = 0;
#pragma unroll
  for (int d = 1; d <= 3; ++d) {
#pragma unroll
    for (int j = 0; j + d < NK; ++j)
      iv[idx++] = 1.0f / (k[j + d] - k[j] + EPSV);   // 11 + 10 + 9 = 30
  }
}

// ---------------------------------------------------------------------------
// Kernel 2: per token: x transpose, activation matrix A[bt][i*4+a], gates.
// ---------------------------------------------------------------------------
__global__ void kan_prep(const float* __restrict__ x,
                         const float* __restrict__ sel_w,
                         const float* __restrict__ sel_b,
                         float* __restrict__ xT,
                         float* __restrict__ acts,
                         float* __restrict__ gate) {
  int t = blockIdx.x;      // token 0..2047
  int i = threadIdx.x;     // 0..63
  __shared__ float sx[INDIM];
  float xv = x[t * INDIM + i];
  sx[i] = xv;
  xT[i * BT + t] = xv;
  float th = tanhf(xv);
  float sg = 1.0f / (1.0f + expf(-xv));
  reinterpret_cast<float4*>(acts)[t * INDIM + i] =
      make_float4(fmaxf(xv, 0.0f), th, sg, xv);
  __syncthreads();
  if (i < NS) {
    float acc = sel_b[i];
#pragma unroll 8
    for (int j = 0; j < INDIM; ++j) acc += sx[j] * sel_w[i * INDIM + j];
    gate[t * NS + i] = 1.0f / (1.0f + expf(-acc));
  }
}

// ---------------------------------------------------------------------------
// Kernel 3: main spline contraction. Block = 256 tokens x one output o.
// Knots / inv-denoms / control points for this o staged in LDS (12.8 KB),
// broadcast-read by all lanes. Fully unrolled Cox-de Boor, degree 3.
// Writes d_out (overwrites poison).
// ---------------------------------------------------------------------------
__global__ void kan_spline(const float* __restrict__ xT,
                           const float* __restrict__ knots,
                           const float* __restrict__ invd,
                           const float* __restrict__ gate,
                           const float* __restrict__ cp,
                           float* __restrict__ out) {
  const int o   = blockIdx.y;
  const int tid = threadIdx.x;
  const int t   = blockIdx.x * 256 + tid;

  __shared__ float s_k [INDIM * NK];   // 768 floats
  __shared__ float s_iv[INDIM * 30];   // 1920 floats
  __shared__ float s_cp[INDIM * NS];   // 512 floats
  for (int idx = tid; idx < INDIM * NK; idx += 256) s_k [idx] = knots[o * INDIM * NK + idx];
  for (int idx = tid; idx < INDIM * 30; idx += 256) s_iv[idx] = invd [o * INDIM * 30 + idx];
  for (int idx = tid; idx < INDIM * NS; idx += 256) s_cp[idx] = cp   [o * INDIM * NS + idx];
  __syncthreads();

  float g[NS];
#pragma unroll
  for (int s = 0; s < NS; ++s) g[s] = gate[t * NS + s];

  float acc = 0.0f;
  for (int i = 0; i < INDIM; ++i) {
    const float xv = xT[i * BT + t];
    const float* kp = &s_k [i * NK];
    const float* iv = &s_iv[i * 30];
    float kk[NK];
#pragma unroll
    for (int j = 0; j < NK; ++j) kk[j] = kp[j];

    float b0[NK - 1];
#pragma unroll
    for (int j = 0; j < NK - 1; ++j)
      b0[j] = (xv >= kk[j] && xv < kk[j + 1]) ? 1.0f : 0.0f;

    float b1[10];
#pragma unroll
    for (int j = 0; j < 10; ++j)
      b1[j] = (xv - kk[j]) * iv[j] * b0[j] + (kk[j + 2] - xv) * iv[j + 1] * b0[j + 1];

    float b2[9];
#pragma unroll
    for (int j = 0; j < 9; ++j)
      b2[j] = (xv - kk[j]) * iv[11 + j] * b1[j] + (kk[j + 3] - xv) * iv[12 + j] * b1[j + 1];

    float b3[NS];
#pragma unroll
    for (int j = 0; j < NS; ++j)
      b3[j] = (xv - kk[j]) * iv[21 + j] * b2[j] + (kk[j + 4] - xv) * iv[22 + j] * b2[j + 1];

    const float* c = &s_cp[i * NS];
#pragma unroll
    for (int s = 0; s < NS; ++s) acc += b3[s] * g[s] * c[s];
  }
  out[t * OUTDIM + o] = acc;
}

// ---------------------------------------------------------------------------
// Kernel 4: activation GEMM (2048x256 @ 256x64) via V_WMMA_F32_16X16X4_F32.
// One wave32 per 16x16 output tile, K-loop of 64 WMMAs.
// A frag:  lane L holds A[M=L%16, K=2*(L/16)+v], v=0,1  (f32 16x4 layout)
// B frag:  lane L holds B[K=2*(L/16)+v, N=L%16]; B[k][n] = aw[n*256+k]
// D:       VGPR v, lane L -> row M = v + 8*(L/16), col N = L%16
// Accumulates 0.1*result into d_out (after kan_spline, stream-ordered).
// ---------------------------------------------------------------------------
__global__ void kan_act_wmma(const float* __restrict__ acts,
                             const float* __restrict__ aw,
                             float* __restrict__ out) {
  const int wave  = (blockIdx.x * blockDim.x + threadIdx.x) >> 5;  // 0..511
  const int lane  = threadIdx.x & 31;
  const int tileM = wave & 127;     // 128 token tiles
  const int tileN = wave >> 7;      // 4 output tiles
  const int r     = lane & 15;
  const int khalf = (lane >> 4) << 1;      // 0 or 2

  const float* aRow = acts + (size_t)(tileM * 16 + r) * KDIM + khalf;
  const float* bRow = aw   + (size_t)(tileN * 16 + r) * KDIM + khalf;

  v8f acc = {0.f, 0.f, 0.f, 0.f, 0.f, 0.f, 0.f, 0.f};

#if defined(__HIP_DEVICE_COMPILE__) && __has_builtin(__builtin_amdgcn_wmma_f32_16x16x4_f32)
#pragma unroll 8
  for (int k = 0; k < KDIM; k += 4) {
    v2f a = *reinterpret_cast<const v2f*>(aRow + k);
    v2f b = *reinterpret_cast<const v2f*>(bRow + k);
    acc = __builtin_amdgcn_wmma_f32_16x16x4_f32(
        /*neg_a=*/false, a, /*neg_b=*/false, b,
        /*c_mod=*/(short)0, acc, /*reuse_a=*/false, /*reuse_b=*/false);
  }
#else
  // scalar fallback producing the identical per-lane D fragment
  {
    const int mBase = (lane >> 4) * 8;
    const int n     = tileN * 16 + (lane & 15);
#pragma unroll
    for (int v = 0; v < 8; ++v) {
      const int m = tileM * 16 + mBase + v;
      float s = 0.0f;
      for (int k = 0; k < KDIM; ++k)
        s += acts[(size_t)m * KDIM + k] * aw[(size_t)n * KDIM + k];
      acc[v] = s;
    }
  }
#endif

  const int mBase  = (lane >> 4) * 8;
  const int col    = lane & 15;
  const int token0 = tileM * 16;
  const int o      = tileN * 16 + col;
#pragma unroll
  for (int v = 0; v < 8; ++v) {
    const int token = token0 + mBase + v;
    out[token * OUTDIM + o] += 0.1f * acc[v];
  }
}

// ---------------------------------------------------------------------------
// Workspace layout (floats):
//   knots: 4096*12 = 49152      @ 0
//   invd : 4096*30 = 122880     @ 49152
//   gate : 2048*8  = 16384      @ 172032
//   xT   : 64*2048 = 131072     @ 188416
//   acts : 2048*256= 524288     @ 319488   (16B-aligned for float4)
// total ~3.3 MB
// ---------------------------------------------------------------------------
extern "C" void kernel_launch(void* const* d_in, const int* in_sizes, int n_in,
                              void* d_out, int out_size, void* d_ws, size_t ws_size,
                              hipStream_t stream) {
  (void)in_sizes; (void)n_in; (void)out_size; (void)ws_size;
  const float* x    = (const float*)d_in[0];
  const float* cp   = (const float*)d_in[1];
  const float* rawk = (const float*)d_in[2];
  const float* aw   = (const float*)d_in[3];
  const float* selw = (const float*)d_in[4];
  const float* selb = (const float*)d_in[5];
  float* out = (float*)d_out;

  float* ws       = (float*)d_ws;
  float* ws_knots = ws;
  float* ws_invd  = ws + 49152;
  float* ws_gate  = ws + 172032;
  float* ws_xT    = ws + 188416;
  float* ws_acts  = ws + 319488;

  kan_sort_knots<<<16, 256, 0, stream>>>(rawk, ws_knots, ws_invd);
  kan_prep<<<BT, INDIM, 0, stream>>>(x, selw, selb, ws_xT, ws_acts, ws_gate);
  kan_spline<<<dim3(BT / 256, OUTDIM), 256, 0, stream>>>(
      ws_xT, ws_knots, ws_invd, ws_gate, cp, out);
  kan_act_wmma<<<64, 256, 0, stream>>>(ws_acts, aw, out);
}